// SmMILPooling_40845138985520
// MI455X (gfx1250) — compile-verified
//
#include <hip/hip_runtime.h>
#include <hip/hip_bf16.h>

#define IN_F   128
#define HID    128
#define NBAGS  128
#define BAG    8192
#define NTOT   (NBAGS * BAG)
#define GRID1  1024      // grid-stride blocks for the WMMA kernel
#define SPLIT  8         // ctx blocks per bag
#define CHUNK  (BAG / SPLIT)

typedef float v2f __attribute__((ext_vector_type(2)));
typedef float v8f __attribute__((ext_vector_type(8)));

// ---------------------------------------------------------------------------
// Kernel 1: logits[i] = tanh(x[i] @ W1 + b1) @ W2 + b2
// Block = 256 threads (8 waves). Each wave computes one 16-row strip with
// v_wmma_f32_16x16x4_f32 chained over K=128 (32 steps) for all 8 column
// tiles (256 WMMAs per strip). Single-strip keeps ~160 live VGPRs -> no
// spills and good occupancy. W1 (64 KB) staged in LDS once per block and
// amortized over 8 grid-stride strips.
// ---------------------------------------------------------------------------
__global__ __launch_bounds__(256)
void k_logits(const float* __restrict__ x, const float* __restrict__ W1,
              const float* __restrict__ b1, const float* __restrict__ W2,
              const float* __restrict__ b2, float* __restrict__ logits)
{
    __shared__ float sW1[IN_F * HID];   // 64 KB, W1[k*HID + h] row-major [K,H]

    const int tid = threadIdx.x;
    for (int i = tid; i < IN_F * HID; i += 256) sW1[i] = W1[i];
    __syncthreads();

    const int wave = tid >> 5;
    const int lane = tid & 31;
    const int half = lane >> 4;     // A/B layout: lanes 0-15 -> K%4 in {0,1}
    const int l16  = lane & 15;
    const float b2v = b2[0];

    const long totalStrips = NTOT / 128;          // 128 rows per block-iter
    for (long strip = blockIdx.x; strip < totalStrips; strip += GRID1) {
        const long rowbase = strip * 128 + (long)wave * 16;
        const long row     = rowbase + l16;       // A-matrix: M = lane&15
        const float* xr    = x + row * IN_F;

        // A tiles: 32 chunks of K=4; per lane a float2 = (K = 4c+2*half, +1)
        v2f a[32];
        #pragma unroll
        for (int c = 0; c < 32; ++c) {
            const int k0 = 4 * c + 2 * half;
            a[c] = *(const v2f*)(xr + k0);        // global_load_b64
        }

        // 8 accumulator tiles: hidden columns [16t, 16t+16)
        v8f acc[8];
        #pragma unroll
        for (int t = 0; t < 8; ++t) acc[t] = (v8f){0.f,0.f,0.f,0.f,0.f,0.f,0.f,0.f};

        #pragma unroll
        for (int c = 0; c < 32; ++c) {
            const int k0 = 4 * c + 2 * half;
            #pragma unroll
            for (int t = 0; t < 8; ++t) {
                const int col = 16 * t + l16;     // B-matrix: N = lane&15
                v2f b;
                b.x = sW1[k0 * HID + col];
                b.y = sW1[(k0 + 1) * HID + col];
                acc[t] = __builtin_amdgcn_wmma_f32_16x16x4_f32(
                    /*neg_a=*/false, a[c], /*neg_b=*/false, b,
                    /*c_mod=*/(short)0, acc[t],
                    /*reuse_a=*/false, /*reuse_b=*/false);
            }
        }

        // Epilogue: h = tanh(acc + b1[col]); partial[v] += h * W2[col]
        // C/D layout: lane L, vgpr v -> row = v + 8*(L>=16), col = (L&15)+16t
        float partial[8];
        #pragma unroll
        for (int v = 0; v < 8; ++v) partial[v] = 0.f;

        #pragma unroll
        for (int t = 0; t < 8; ++t) {
            const int h    = 16 * t + l16;
            const float bb = b1[h];
            const float w2 = W2[h];
            #pragma unroll
            for (int v = 0; v < 8; ++v)
                partial[v] += tanhf(acc[t][v] + bb) * w2;
        }

        // Reduce across the 16 lanes of each half (masks < 16 stay in-half)
        #pragma unroll
        for (int v = 0; v < 8; ++v) {
            float p = partial[v];
            p += __shfl_xor(p, 1, 32);
            p += __shfl_xor(p, 2, 32);
            p += __shfl_xor(p, 4, 32);
            p += __shfl_xor(p, 8, 32);
            partial[v] = p;
        }

        if (l16 == 0) {
            const long r0 = rowbase + 8 * half;   // lane 0 -> rows v, lane 16 -> v+8
            #pragma unroll
            for (int v = 0; v < 8; ++v)
                logits[r0 + v] = partial[v] + b2v;    // TEMP == 1
        }
    }
}

// ---------------------------------------------------------------------------
// Kernel 2: per-bag max, sum(exp), and renorm-sum of the smoothed weights.
// One block per bag. gs = sum_i g_i where g = 0.5*w + 0.5*Aw (pre-renorm).
// ---------------------------------------------------------------------------
__global__ __launch_bounds__(256)
void k_bagstats(const float* __restrict__ logits, float* __restrict__ m,
                float* __restrict__ s, float* __restrict__ gs)
{
    __shared__ float red[256];
    const int bag = blockIdx.x, tid = threadIdx.x;
    const float* lg = logits + (long)bag * BAG;

    float mx = -__builtin_inff();
    for (int i = tid; i < BAG; i += 256) mx = fmaxf(mx, lg[i]);
    red[tid] = mx; __syncthreads();
    for (int st = 128; st > 0; st >>= 1) {
        if (tid < st) red[tid] = fmaxf(red[tid], red[tid + st]);
        __syncthreads();
    }
    const float bm = red[0];
    __syncthreads();

    float sm = 0.f;
    for (int i = tid; i < BAG; i += 256) sm += __expf(lg[i] - bm);
    red[tid] = sm; __syncthreads();
    for (int st = 128; st > 0; st >>= 1) {
        if (tid < st) red[tid] += red[tid + st];
        __syncthreads();
    }
    const float bs = red[0];
    const float inv_s = 1.0f / bs;
    __syncthreads();

    // sum of smoothed (pre-renormalization) weights
    float psum = 0.f;
    for (int i = tid; i < BAG; i += 256) {
        const float w  = __expf(lg[i] - bm) * inv_s;
        const int sl = (i > 0), sr = (i < BAG - 1);
        const float wl = sl ? __expf(lg[i - 1] - bm) * inv_s : 0.f;
        const float wr = sr ? __expf(lg[i + 1] - bm) * inv_s : 0.f;
        const float deg = 1.f + (float)sl + (float)sr;
        psum += 0.5f * w + 0.5f * (w + wl + wr) / deg;
    }
    red[tid] = psum; __syncthreads();
    for (int st = 128; st > 0; st >>= 1) {
        if (tid < st) red[tid] += red[tid + st];
        __syncthreads();
    }
    if (tid == 0) { m[bag] = bm; s[bag] = bs; gs[bag] = red[0]; }
}

// ---------------------------------------------------------------------------
// Kernel 3: per-slice g computation + write + partial ctx reduction.
// SPLIT blocks per bag (1024 total) so the 512 MB x re-read saturates HBM.
// ---------------------------------------------------------------------------
__global__ __launch_bounds__(256)
void k_ctx(const float* __restrict__ x, const float* __restrict__ logits,
           const float* __restrict__ m, const float* __restrict__ s,
           const float* __restrict__ gs, float* __restrict__ gout,
           float* __restrict__ ctxpart)
{
    __shared__ float g[CHUNK];   // 4 KB
    __shared__ float red[256];
    const int tid  = threadIdx.x;
    const int bag  = blockIdx.x / SPLIT;
    const int part = blockIdx.x % SPLIT;
    const int i0   = part * CHUNK;

    const float* lg = logits + (long)bag * BAG;
    const float bm     = m[bag];
    const float inv_s  = 1.0f / s[bag];
    const float inv_gs = 1.0f / (gs[bag] + 1e-8f);

    for (int ii = tid; ii < CHUNK; ii += 256) {
        const int i = i0 + ii;
        const float w  = __expf(lg[i] - bm) * inv_s;
        const int sl = (i > 0), sr = (i < BAG - 1);
        const float wl = sl ? __expf(lg[i - 1] - bm) * inv_s : 0.f;
        const float wr = sr ? __expf(lg[i + 1] - bm) * inv_s : 0.f;
        const float deg = 1.f + (float)sl + (float)sr;
        const float gn = (0.5f * w + 0.5f * (w + wl + wr) / deg) * inv_gs;
        g[ii] = gn;
        gout[(long)bag * BAG + i] = gn;
    }
    __syncthreads();

    // partial ctx over this slice; lane -> feature gives coalesced 128B/wave
    const int f = tid & 127, grp = tid >> 7;
    const float* xb = x + ((long)bag * BAG + i0) * IN_F;
    float acc = 0.f;
    for (int i = grp; i < CHUNK; i += 2) {
        __builtin_prefetch(xb + (long)(i + 16) * IN_F + f, 0, 1); // global_prefetch_b8
        acc += g[i] * xb[(long)i * IN_F + f];
    }
    red[tid] = acc; __syncthreads();
    if (tid < 128)
        ctxpart[(long)blockIdx.x * 128 + tid] = red[tid] + red[tid + 128];
}

// ---------------------------------------------------------------------------
// Kernel 4: combine SPLIT partials per bag into ctx.
// ---------------------------------------------------------------------------
__global__ __launch_bounds__(256)
void k_ctx_final(const float* __restrict__ ctxpart, float* __restrict__ ctx)
{
    const int idx = blockIdx.x * 256 + threadIdx.x;  // NBAGS*128 = 16384
    if (idx >= NBAGS * 128) return;
    const int bag = idx >> 7, f = idx & 127;
    float acc = 0.f;
    #pragma unroll
    for (int p = 0; p < SPLIT; ++p)
        acc += ctxpart[((long)bag * SPLIT + p) * 128 + f];
    ctx[idx] = acc;
}

// ---------------------------------------------------------------------------
extern "C" void kernel_launch(void* const* d_in, const int* in_sizes, int n_in,
                              void* d_out, int out_size, void* d_ws, size_t ws_size,
                              hipStream_t stream)
{
    const float* x  = (const float*)d_in[0];
    const float* W1 = (const float*)d_in[1];
    const float* b1 = (const float*)d_in[2];
    const float* W2 = (const float*)d_in[3];
    const float* b2 = (const float*)d_in[4];
    // d_in[5] = lengths (all == BAG, bags contiguous) -> not needed

    float* ctx  = (float*)d_out;              // [NBAGS, 128]
    float* gout = ctx + NBAGS * IN_F;         // [N, 1]

    float* logits  = (float*)d_ws;            // NTOT floats (4 MB)
    float* m       = logits + NTOT;           // NBAGS
    float* s       = m + NBAGS;               // NBAGS
    float* gs      = s + NBAGS;               // NBAGS
    float* ctxpart = gs + NBAGS;              // NBAGS*SPLIT*128 (512 KB)

    k_logits   <<<GRID1, 256, 0, stream>>>(x, W1, b1, W2, b2, logits);
    k_bagstats <<<NBAGS, 256, 0, stream>>>(logits, m, s, gs);
    k_ctx      <<<NBAGS * SPLIT, 256, 0, stream>>>(x, logits, m, s, gs, gout, ctxpart);
    k_ctx_final<<<(NBAGS * 128 + 255) / 256, 256, 0, stream>>>(ctxpart, ctx);
}